// GAT_21766894256417
// MI455X (gfx1250) — compile-verified
//
#include <hip/hip_runtime.h>
#include <math.h>

typedef float v2f __attribute__((ext_vector_type(2)));
typedef float v8f __attribute__((ext_vector_type(8)));

#define F 128   // feature dim for every matrix in this model

// ---------------------------------------------------------------------------
// Native gfx1250 f32 atomics (non-returning, device scope -> L2 atomic units).
// ---------------------------------------------------------------------------
__device__ __forceinline__ void atomicMaxF32(float* p, float v) {
  asm volatile("global_atomic_max_num_f32 %0, %1, off scope:SCOPE_DEV"
               :: "v"((unsigned long long)(uintptr_t)p), "v"(v) : "memory");
}
__device__ __forceinline__ void atomicAddF32(float* p, float v) {
  asm volatile("global_atomic_add_f32 %0, %1, off scope:SCOPE_DEV"
               :: "v"((unsigned long long)(uintptr_t)p), "v"(v) : "memory");
}

// ---------------------------------------------------------------------------
// GEMM: Y[n x 128] = X[n x 128] @ W[128 x 128], f32 WMMA 16x16x4.
// Block = 256 threads = 8 waves; block covers 8 row tiles (128 rows).
// W (64KB) is staged once per block into LDS via async copy (ASYNCcnt),
// XOR-swizzled so the half-wave K-split reads are LDS-bank conflict free.
// Each wave owns one row tile and sweeps all 8 column tiles, reusing its
// A-fragment across 8 WMMAs per K step.
// ---------------------------------------------------------------------------
__device__ __forceinline__ int swz(int r, int c) {      // LDS float index
  return r * F + (c ^ (((r >> 1) & 1) << 5));
}

__global__ __launch_bounds__(256) void gatv2_gemm128_wmma(
    const float* __restrict__ X, const float* __restrict__ W,
    float* __restrict__ Y, int rowTiles) {
  __shared__ float sW[F * F];                     // 64 KB
  const int tid = threadIdx.x;

  // ---- async stage W -> LDS: 4096 chunks of 16B, swizzled placement ----
  {
    unsigned ldsbase = (unsigned)(uintptr_t)(&sW[0]);
    unsigned long long gbase = (unsigned long long)(uintptr_t)W;
#pragma unroll
    for (int i = 0; i < 16; ++i) {
      unsigned cidx = (unsigned)tid + 256u * i;   // 16B chunk id
      unsigned row  = cidx >> 5;                  // 32 chunks per 512B row
      unsigned cb   = (cidx & 31u) * 16u;         // byte offset in row
      unsigned lds  = ldsbase + row * 512u + (cb ^ (((row >> 1) & 1u) << 7));
      unsigned gof  = cidx * 16u;
      asm volatile("global_load_async_to_lds_b128 %0, %1, %2"
                   :: "v"(lds), "v"(gof), "s"(gbase) : "memory");
    }
    asm volatile("s_wait_asynccnt 0" ::: "memory");
  }
  __syncthreads();

  const int wave = tid >> 5;
  const int lane = tid & 31;
  const int m16  = lane & 15;
  const int kb   = (lane >> 4) * 2;               // K sub-offset: 0 or 2
  const int rt   = blockIdx.x * 8 + wave;         // this wave's row tile

  if (rt < rowTiles) {                            // uniform per wave
    const int row = rt * 16 + m16;
    v8f acc[8];
#pragma unroll
    for (int ct = 0; ct < 8; ++ct) acc[ct] = (v8f){};

    for (int k = 0; k < F; k += 4) {
      v2f a;
      a.x = X[row * F + (k + kb)];
      a.y = X[row * F + (k + kb + 1)];
#pragma unroll
      for (int ct = 0; ct < 8; ++ct) {
        v2f b;
        const int col = ct * 16 + m16;
        b.x = sW[swz(k + kb, col)];
        b.y = sW[swz(k + kb + 1, col)];
        acc[ct] = __builtin_amdgcn_wmma_f32_16x16x4_f32(
            false, a, false, b, (short)0, acc[ct], false, false);
      }
    }

    const int mbase = (lane >> 4) * 8;            // C/D: lanes 16-31 -> M=8..15
#pragma unroll
    for (int ct = 0; ct < 8; ++ct) {
#pragma unroll
      for (int v = 0; v < 8; ++v) {
        Y[(rt * 16 + mbase + v) * F + ct * 16 + m16] = acc[ct][v];
      }
    }
  }
}

// ---------------------------------------------------------------------------
// helpers
// ---------------------------------------------------------------------------
__global__ void fill_f32(float* __restrict__ p, float v, int nelts) {
  int i = blockIdx.x * blockDim.x + threadIdx.x;
  if (i < nelts) p[i] = v;
}

__global__ void bias_relu_k(float* __restrict__ h, const float* __restrict__ b,
                            int nelts) {
  int i = blockIdx.x * blockDim.x + threadIdx.x;
  if (i < nelts) {
    float v = h[i] + b[i & (F - 1)];
    h[i] = v > 0.f ? v : 0.f;
  }
}

__global__ void bias_add_k(float* __restrict__ o, const float* __restrict__ b,
                           int nelts) {
  int i = blockIdx.x * blockDim.x + threadIdx.x;
  if (i < nelts) o[i] += b[i & (F - 1)];
}

// ---------------------------------------------------------------------------
// Edge pass 1: raw attention logits + segment max.
// One wave32 per edge; lane j handles features j, j+32, j+64, j+96 (coalesced
// 128B bursts). H in {4,1}; C = 128/H. For H=4 chunk i == head i.
// Self-loops are implicit: edge id >= E  ->  src = dst = id - E.
// ---------------------------------------------------------------------------
template <int H>
__global__ void edge_alpha_k(const long long* __restrict__ src,
                             const long long* __restrict__ dst,
                             const float* __restrict__ xl,
                             const float* __restrict__ xr,
                             const float* __restrict__ att,
                             float* __restrict__ alphaRaw,
                             float* __restrict__ segMax, int E, int Etot) {
  const int e    = (blockIdx.x * blockDim.x + threadIdx.x) >> 5;
  const int lane = threadIdx.x & 31;
  if (e >= Etot) return;
  int s, d;
  if (e < E) { s = (int)src[e]; d = (int)dst[e]; } else { s = d = e - E; }

  float p[4];
#pragma unroll
  for (int i = 0; i < 4; ++i) {
    const int j = lane + 32 * i;
    float v = xl[s * F + j] + xr[d * F + j];
    v = v > 0.f ? v : 0.2f * v;          // leaky_relu, slope 0.2
    p[i] = v * att[j];
  }
  if (H == 1) p[0] = p[0] + p[1] + p[2] + p[3];

#pragma unroll
  for (int off = 16; off > 0; off >>= 1) {
    if (H == 4) {
#pragma unroll
      for (int i = 0; i < 4; ++i) p[i] += __shfl_down(p[i], off, 32);
    } else {
      p[0] += __shfl_down(p[0], off, 32);
    }
  }
  if (lane == 0) {
#pragma unroll
    for (int h = 0; h < H; ++h) {
      const float a = (H == 4) ? p[h] : p[0];
      alphaRaw[(size_t)e * H + h] = a;
      atomicMaxF32(&segMax[d * H + h], a);
    }
  }
}

// ---------------------------------------------------------------------------
// Edge pass 2: alpha = exp(raw - segmax[dst]); accumulate denominator.
// One thread per (edge, head).
// ---------------------------------------------------------------------------
template <int H>
__global__ void edge_exp_k(const long long* __restrict__ dst,
                           const float* __restrict__ segMax,
                           float* __restrict__ alpha,
                           float* __restrict__ segSum, int E, int Etot) {
  int t = blockIdx.x * blockDim.x + threadIdx.x;
  if (t >= Etot * H) return;
  const int e = t / H, h = t - e * H;
  const int d = (e < E) ? (int)dst[e] : e - E;
  const float a = __expf(alpha[t] - segMax[d * H + h]);
  alpha[t] = a;
  atomicAddF32(&segSum[d * H + h], a);
}

// ---------------------------------------------------------------------------
// Edge pass 3: out[dst] += (alpha/denom) * x_l[src].  One wave per edge.
// ---------------------------------------------------------------------------
template <int H>
__global__ void edge_scatter_k(const long long* __restrict__ src,
                               const long long* __restrict__ dst,
                               const float* __restrict__ xl,
                               const float* __restrict__ alpha,
                               const float* __restrict__ segSum,
                               float* __restrict__ out, int E, int Etot) {
  const int e    = (blockIdx.x * blockDim.x + threadIdx.x) >> 5;
  const int lane = threadIdx.x & 31;
  if (e >= Etot) return;
  int s, d;
  if (e < E) { s = (int)src[e]; d = (int)dst[e]; } else { s = d = e - E; }

  float w[H];
#pragma unroll
  for (int h = 0; h < H; ++h)
    w[h] = alpha[(size_t)e * H + h] / segSum[d * H + h];

#pragma unroll
  for (int i = 0; i < 4; ++i) {
    const int j = lane + 32 * i;
    const int h = (H == 4) ? i : 0;
    atomicAddF32(&out[d * F + j], w[h] * xl[s * F + j]);
  }
}

// ---------------------------------------------------------------------------
// launch
// ---------------------------------------------------------------------------
extern "C" void kernel_launch(void* const* d_in, const int* in_sizes, int n_in,
                              void* d_out, int out_size, void* d_ws,
                              size_t ws_size, hipStream_t stream) {
  const float*     x    = (const float*)d_in[0];
  const long long* ei   = (const long long*)d_in[1];   // int64 in reference
  const float*     Wl1  = (const float*)d_in[2];
  const float*     Wr1  = (const float*)d_in[3];
  const float*     att1 = (const float*)d_in[4];       // [4,32] flat = 128
  const float*     b1   = (const float*)d_in[5];
  const float*     Wl2  = (const float*)d_in[6];
  const float*     Wr2  = (const float*)d_in[7];
  const float*     att2 = (const float*)d_in[8];       // [1,128] flat = 128
  const float*     b2   = (const float*)d_in[9];

  const int n    = in_sizes[0] / F;   // 50000
  const int E    = in_sizes[1] / 2;   // 800000
  const int Etot = E + n;             // with self-loops
  const long long* src = ei;
  const long long* dst = ei + E;
  float* out = (float*)d_out;

  // workspace carve (floats): xl | xr | h | alpha | segMax | segSum
  float* ws = (float*)d_ws;
  size_t o = 0;
  float* xl     = ws + o; o += (size_t)n * F;
  float* xr     = ws + o; o += (size_t)n * F;
  float* h      = ws + o; o += (size_t)n * F;
  float* alpha  = ws + o; o += (size_t)Etot * 4;
  float* segMax = ws + o; o += (size_t)n * 4;
  float* segSum = ws + o; o += (size_t)n * 4;

  const int rowTiles   = n / 16;                 // 3125, exact
  const int gemmBlocks = (rowTiles + 7) / 8;     // 8 row tiles per block
  const int edgeBlocks = (Etot + 7) / 8;         // 8 waves / block
  const dim3 blk(256);

  // ---------------- layer 1: GATv2(128 -> 4 heads x 32) ----------------
  gatv2_gemm128_wmma<<<gemmBlocks, blk, 0, stream>>>(x, Wl1, xl, rowTiles);
  gatv2_gemm128_wmma<<<gemmBlocks, blk, 0, stream>>>(x, Wr1, xr, rowTiles);
  fill_f32<<<(n * 4 + 255) / 256, blk, 0, stream>>>(segMax, -3.0e38f, n * 4);
  fill_f32<<<(n * 4 + 255) / 256, blk, 0, stream>>>(segSum, 0.f, n * 4);
  fill_f32<<<(n * F + 255) / 256, blk, 0, stream>>>(h, 0.f, n * F);
  edge_alpha_k<4><<<edgeBlocks, blk, 0, stream>>>(src, dst, xl, xr, att1,
                                                  alpha, segMax, E, Etot);
  edge_exp_k<4><<<(Etot * 4 + 255) / 256, blk, 0, stream>>>(dst, segMax, alpha,
                                                            segSum, E, Etot);
  edge_scatter_k<4><<<edgeBlocks, blk, 0, stream>>>(src, dst, xl, alpha,
                                                    segSum, h, E, Etot);
  bias_relu_k<<<(n * F + 255) / 256, blk, 0, stream>>>(h, b1, n * F);

  // ---------------- layer 2: GATv2(128 -> 1 head x 128) ----------------
  gatv2_gemm128_wmma<<<gemmBlocks, blk, 0, stream>>>(h, Wl2, xl, rowTiles);
  gatv2_gemm128_wmma<<<gemmBlocks, blk, 0, stream>>>(h, Wr2, xr, rowTiles);
  fill_f32<<<(n + 255) / 256, blk, 0, stream>>>(segMax, -3.0e38f, n);
  fill_f32<<<(n + 255) / 256, blk, 0, stream>>>(segSum, 0.f, n);
  fill_f32<<<(n * F + 255) / 256, blk, 0, stream>>>(out, 0.f, n * F);
  edge_alpha_k<1><<<edgeBlocks, blk, 0, stream>>>(src, dst, xl, xr, att2,
                                                  alpha, segMax, E, Etot);
  edge_exp_k<1><<<(Etot + 255) / 256, blk, 0, stream>>>(dst, segMax, alpha,
                                                        segSum, E, Etot);
  edge_scatter_k<1><<<edgeBlocks, blk, 0, stream>>>(src, dst, xl, alpha,
                                                    segSum, out, E, Etot);
  bias_add_k<<<(n * F + 255) / 256, blk, 0, stream>>>(out, b2, n * F);
}